// MSC_Cell_64656437674476
// MI455X (gfx1250) — compile-verified
//
#include <hip/hip_runtime.h>
#include <math.h>

#define STATE_DIM 5
#define HIDDEN 32
#define B_DIM 128
#define S_DIM 4096
#define NPOS (B_DIM * S_DIM)   /* 524288 */
#define D_IN 9
#define KPAD0 12

typedef __attribute__((ext_vector_type(2))) float v2f;
typedef __attribute__((ext_vector_type(8))) float v8f;

__device__ __forceinline__ v8f wmma4(v2f a, v2f b, v8f c) {
  // D = A(16x4,f32) x B(4x16,f32) + C(16x16,f32)
  return __builtin_amdgcn_wmma_f32_16x16x4_f32(false, a, false, b, (short)0, c,
                                               false, false);
}

__device__ __forceinline__ v2f make2(float x, float y) {
  v2f v; v[0] = x; v[1] = y; return v;
}

#define LOG2E 1.4426950408889634f

// branch-free hardware transcendentals: v_exp_f32 / v_rcp_f32
__device__ __forceinline__ float fast_exp(float x) {
  return __builtin_amdgcn_exp2f(x * LOG2E);
}
__device__ __forceinline__ float fast_tanh(float x) {
  const float ax = fabsf(x);
  const float u = __builtin_amdgcn_exp2f(ax * (-2.f * LOG2E));  // e^(-2|x|)
  const float t = (1.f - u) * __builtin_amdgcn_rcpf(1.f + u);
  return copysignf(t, x);
}

// ---------------------------------------------------------------------------
// Kernel 1: temp_seq[b,s] = delta_x[b,0,5] + inclusive_cumsum_s(delta_x[b,s,2])
// One block (256 threads) per batch row; each thread owns 16 contiguous s.
// ---------------------------------------------------------------------------
__global__ __launch_bounds__(256) void temp_scan_kernel(
    const float* __restrict__ delta_x, float* __restrict__ temp) {
  const int b = blockIdx.x;
  const int t = threadIdx.x;
  const int ELEMS = S_DIM / 256;  // 16
  const float init = delta_x[((size_t)b * S_DIM) * 6 + 5];

  float v[16];
  float run = 0.f;
  const size_t base = (size_t)b * S_DIM + (size_t)t * ELEMS;
#pragma unroll
  for (int i = 0; i < ELEMS; ++i) {
    run += delta_x[(base + i) * 6 + 2];
    v[i] = run;  // local inclusive prefix
  }

  __shared__ float ssum[256];
  ssum[t] = run;
  __syncthreads();
  for (int off = 1; off < 256; off <<= 1) {
    float add = (t >= off) ? ssum[t - off] : 0.f;
    __syncthreads();
    ssum[t] += add;
    __syncthreads();
  }
  const float excl = ssum[t] - run;  // exclusive prefix of thread sums
#pragma unroll
  for (int i = 0; i < ELEMS; ++i) temp[base + i] = init + excl + v[i];
}

// ---------------------------------------------------------------------------
// Kernel 2: fused MLP via fp32 WMMA. 1 wave = 16 positions. 4 waves / block.
// ---------------------------------------------------------------------------
__global__ __launch_bounds__(128) void msc_wmma_kernel(
    const float* __restrict__ h_prev, const float* __restrict__ delta_x,
    const float* __restrict__ Wa0, const float* __restrict__ ba0,
    const float* __restrict__ Wb0, const float* __restrict__ bb0,
    const float* __restrict__ Wa1, const float* __restrict__ ba1,
    const float* __restrict__ Wb1, const float* __restrict__ bb1,
    const float* __restrict__ W_alpha, const float* __restrict__ b_alpha,
    const float* __restrict__ W_beta, const float* __restrict__ b_beta,
    const float* __restrict__ W_gamma, const float* __restrict__ b_gamma,
    const float* __restrict__ W_c, const float* __restrict__ b_c,
    const float* __restrict__ W_out, const float* __restrict__ temp,
    float* __restrict__ out) {
  __shared__ float sWa0[KPAD0 * 32], sWb0[KPAD0 * 32];
  __shared__ float sWa1[32 * 32], sWb1[32 * 32];
  __shared__ float sWh[32 * 16];  // fused head: [alpha|beta|gamma|W_c(5)|pad]
  __shared__ float sBa0[32], sBb0[32], sBa1[32], sBb1[32], sBh[16], sWout[8];
  __shared__ float sAct[4][16 * 32];  // per-wave activation relayout buffer
  __shared__ float sHead[4][16 * 8];  // per-wave head outputs

  const int tid = threadIdx.x;

  // ---- stage weights to LDS (zero-padded where needed) ----
  for (int i = tid; i < KPAD0 * 32; i += 128) {
    int kk = i >> 5, c = i & 31;
    sWa0[i] = (kk < D_IN) ? Wa0[kk * 32 + c] : 0.f;
    sWb0[i] = (kk < D_IN) ? Wb0[kk * 32 + c] : 0.f;
  }
  for (int i = tid; i < 1024; i += 128) {
    sWa1[i] = Wa1[i];
    sWb1[i] = Wb1[i];
  }
  for (int i = tid; i < 512; i += 128) {
    int kk = i >> 4, c = i & 15;
    float v = 0.f;
    if (c == 0)      v = W_alpha[kk];
    else if (c == 1) v = W_beta[kk];
    else if (c == 2) v = W_gamma[kk];
    else if (c < 8)  v = W_c[kk * 5 + (c - 3)];
    sWh[i] = v;
  }
  if (tid < 32) {
    sBa0[tid] = ba0[tid]; sBb0[tid] = bb0[tid];
    sBa1[tid] = ba1[tid]; sBb1[tid] = bb1[tid];
  }
  if (tid < 16) {
    float v = 0.f;
    if (tid == 0)      v = b_alpha[0];
    else if (tid == 1) v = b_beta[0];
    else if (tid == 2) v = b_gamma[0];
    else if (tid < 8)  v = b_c[tid - 3];
    sBh[tid] = v;
  }
  if (tid < 8) sWout[tid] = (tid < 5) ? W_out[tid] : 0.f;
  __syncthreads();

  const int wave = tid >> 5;
  const int lane = tid & 31;
  const int r = lane & 15;   // row within tile (A) / col within tile (B,C)
  const int hi = lane >> 4;  // which K/M half this lane covers
  const int tile = blockIdx.x * 4 + wave;
  const size_t p = (size_t)tile * 16 + r;

  // ---- per-lane inputs for its A row (both lane halves load row r) ----
  float hp[STATE_DIM];
#pragma unroll
  for (int j = 0; j < STATE_DIM; ++j) hp[j] = h_prev[p * STATE_DIM + j];
  const float dx0 = delta_x[p * 6 + 0];
  const float dx1 = delta_x[p * 6 + 1];
  const float dx2 = delta_x[p * 6 + 2];
  const float tse = temp[p];
  float nrm = __builtin_amdgcn_sqrtf(dx0 * dx0 + dx1 * dx1 + dx2 * dx2);
  nrm = fmaxf(nrm, 1e-7f);
  const float inv = __builtin_amdgcn_rcpf(nrm);

  float l0[KPAD0];
  l0[0] = hp[0]; l0[1] = hp[1]; l0[2] = hp[2]; l0[3] = hp[3]; l0[4] = hp[4];
  l0[5] = tse;
  l0[6] = dx0 * inv; l0[7] = dx1 * inv; l0[8] = dx2 * inv;
  l0[9] = 0.f; l0[10] = 0.f; l0[11] = 0.f;

  float* act = sAct[wave];
  v8f accA0, accA1, accB0, accB1;

  // ================= layer 0 : (16x12) @ (12x32), gated tanh =================
  {
    const float bA0 = sBa0[r], bA1 = sBa0[16 + r];
    const float bB0 = sBb0[r], bB1 = sBb0[16 + r];
#pragma unroll
    for (int j = 0; j < 8; ++j) {
      accA0[j] = bA0; accA1[j] = bA1; accB0[j] = bB0; accB1[j] = bB1;
    }
  }
#pragma unroll
  for (int k = 0; k < 3; ++k) {
    const v2f a = make2(hi ? l0[4 * k + 2] : l0[4 * k + 0],
                        hi ? l0[4 * k + 3] : l0[4 * k + 1]);
    const int brow = 4 * k + 2 * hi;
    accA0 = wmma4(a, make2(sWa0[brow * 32 + r],       sWa0[(brow + 1) * 32 + r]),       accA0);
    accA1 = wmma4(a, make2(sWa0[brow * 32 + 16 + r],  sWa0[(brow + 1) * 32 + 16 + r]),  accA1);
    accB0 = wmma4(a, make2(sWb0[brow * 32 + r],       sWb0[(brow + 1) * 32 + r]),       accB0);
    accB1 = wmma4(a, make2(sWb0[brow * 32 + 16 + r],  sWb0[(brow + 1) * 32 + 16 + r]),  accB1);
  }
  // C layout -> row-major 16x32 in LDS (lane holds col N = 16n+r, rows j+8*hi)
#pragma unroll
  for (int j = 0; j < 8; ++j) {
    const int M = j + 8 * hi;
    act[M * 32 + r]      = fast_tanh(accA0[j]) * fast_tanh(accB0[j]);
    act[M * 32 + 16 + r] = fast_tanh(accA1[j]) * fast_tanh(accB1[j]);
  }
  __syncthreads();

  // ================= layer 1 : (16x32) @ (32x32), gated tanh =================
  {
    const float bA0 = sBa1[r], bA1 = sBa1[16 + r];
    const float bB0 = sBb1[r], bB1 = sBb1[16 + r];
#pragma unroll
    for (int j = 0; j < 8; ++j) {
      accA0[j] = bA0; accA1[j] = bA1; accB0[j] = bB0; accB1[j] = bB1;
    }
  }
#pragma unroll
  for (int k = 0; k < 8; ++k) {
    const int arow = r * 32 + 4 * k + 2 * hi;
    const v2f a = make2(act[arow], act[arow + 1]);
    const int brow = 4 * k + 2 * hi;
    accA0 = wmma4(a, make2(sWa1[brow * 32 + r],       sWa1[(brow + 1) * 32 + r]),       accA0);
    accA1 = wmma4(a, make2(sWa1[brow * 32 + 16 + r],  sWa1[(brow + 1) * 32 + 16 + r]),  accA1);
    accB0 = wmma4(a, make2(sWb1[brow * 32 + r],       sWb1[(brow + 1) * 32 + r]),       accB0);
    accB1 = wmma4(a, make2(sWb1[brow * 32 + 16 + r],  sWb1[(brow + 1) * 32 + 16 + r]),  accB1);
  }
  __syncthreads();  // everyone done reading act before overwrite
#pragma unroll
  for (int j = 0; j < 8; ++j) {
    const int M = j + 8 * hi;
    act[M * 32 + r]      = fast_tanh(accA0[j]) * fast_tanh(accB0[j]);
    act[M * 32 + 16 + r] = fast_tanh(accA1[j]) * fast_tanh(accB1[j]);
  }
  __syncthreads();

  // ================= head : (16x32) @ (32x16) fused [a|b|g|c] ================
  v8f accH;
  {
    const float bh = sBh[r];
#pragma unroll
    for (int j = 0; j < 8; ++j) accH[j] = bh;
  }
#pragma unroll
  for (int k = 0; k < 8; ++k) {
    const int arow = r * 32 + 4 * k + 2 * hi;
    const v2f a = make2(act[arow], act[arow + 1]);
    const int brow = 4 * k + 2 * hi;
    accH = wmma4(a, make2(sWh[brow * 16 + r], sWh[(brow + 1) * 16 + r]), accH);
  }
  float* hbuf = sHead[wave];
  if (r < 8) {
#pragma unroll
    for (int j = 0; j < 8; ++j) hbuf[(j + 8 * hi) * 8 + r] = accH[j];
  }
  __syncthreads();

  // ================= pointwise tail (one lane per position) =================
  if (lane < 16) {
    const float alpha = fast_exp(hbuf[lane * 8 + 0]);
    const float beta  = fast_exp(hbuf[lane * 8 + 1]);
    const float gamma = fast_exp(hbuf[lane * 8 + 2]);
    const float z =
        1.f - fast_exp(-alpha * fabsf(dx0) - beta * dx1 - gamma * fabsf(dx2));
    float sigma = 0.f;
#pragma unroll
    for (int j = 0; j < STATE_DIM; ++j) {
      const float cn = fast_tanh(hbuf[lane * 8 + 3 + j]);
      const float hn = (1.f - z) * hp[j] + z * cn;
      out[p * STATE_DIM + j] = hn;
      sigma += hn * sWout[j];
    }
    out[(size_t)NPOS * STATE_DIM + p] = sigma;
  }
}

// ---------------------------------------------------------------------------
extern "C" void kernel_launch(void* const* d_in, const int* in_sizes, int n_in,
                              void* d_out, int out_size, void* d_ws,
                              size_t ws_size, hipStream_t stream) {
  const float* h_prev  = (const float*)d_in[0];
  const float* delta_x = (const float*)d_in[1];
  const float* Wa0 = (const float*)d_in[2];
  const float* ba0 = (const float*)d_in[3];
  const float* Wb0 = (const float*)d_in[4];
  const float* bb0 = (const float*)d_in[5];
  const float* Wa1 = (const float*)d_in[6];
  const float* ba1 = (const float*)d_in[7];
  const float* Wb1 = (const float*)d_in[8];
  const float* bb1 = (const float*)d_in[9];
  const float* W_alpha = (const float*)d_in[10];
  const float* b_alpha = (const float*)d_in[11];
  const float* W_beta  = (const float*)d_in[12];
  const float* b_beta  = (const float*)d_in[13];
  const float* W_gamma = (const float*)d_in[14];
  const float* b_gamma = (const float*)d_in[15];
  const float* W_c   = (const float*)d_in[16];
  const float* b_c   = (const float*)d_in[17];
  const float* W_out = (const float*)d_in[18];

  float* temp = (float*)d_ws;  // NPOS floats = 2 MB
  float* out = (float*)d_out;

  temp_scan_kernel<<<B_DIM, 256, 0, stream>>>(delta_x, temp);

  const int tilesPerBlock = 4;                       // 4 waves/block
  const int nBlocks = NPOS / 16 / tilesPerBlock;     // 8192
  msc_wmma_kernel<<<nBlocks, 128, 0, stream>>>(
      h_prev, delta_x, Wa0, ba0, Wb0, bb0, Wa1, ba1, Wb1, bb1, W_alpha, b_alpha,
      W_beta, b_beta, W_gamma, b_gamma, W_c, b_c, W_out, temp, out);
}